// AMN_18004502905276
// MI455X (gfx1250) — compile-verified
//
#include <hip/hip_runtime.h>

typedef __attribute__((ext_vector_type(2))) float v2f;
typedef __attribute__((ext_vector_type(4))) float v4f;
typedef __attribute__((ext_vector_type(8))) float v8f;

namespace {
constexpr int U_  = 64;
constexpr int T_  = 100;
constexpr int N_  = 4096;
constexpr int TN_ = T_ * N_;                                    // 409600
constexpr int WAVES_PER_BLOCK = 8;                              // 256 threads (wave32)
constexpr int TILES_PER_WAVE  = 4;                              // 4 x 16 outputs per wave
constexpr int OUT_PER_WAVE    = TILES_PER_WAVE * 16;            // 64
constexpr int OUT_PER_BLOCK   = WAVES_PER_BLOCK * OUT_PER_WAVE; // 512
constexpr int GRID_BLOCKS     = TN_ / OUT_PER_BLOCK;            // 800, exact
}

// Step 1: w[j] = 3 * sum_i (conn[i][j] > 0.1 ? conn[i][j] : 0)   -- 64x64, tiny
__global__ void amn_weights_kernel(const float* __restrict__ conn,
                                   float* __restrict__ w) {
    const int j = threadIdx.x;  // launched with 64 threads
    float s = 0.0f;
#pragma unroll
    for (int i = 0; i < U_; ++i) {
        const float c = conn[i * U_ + j];
        s += (c > 0.1f) ? c : 0.0f;
    }
    w[j] = 3.0f * s;
}

// Step 2: out[tn] = sum_j w[j] * X[j][tn], via V_WMMA_F32_16X16X4_F32.
// A tile: 16 tn-positions (M) x 4 units (K) of X.
// B tile: w[k] broadcast across all 16 columns (N) -> every D column identical.
__global__ __launch_bounds__(256) void amn_wsum_wmma_kernel(
        const float* __restrict__ X,    // [U, T*N]
        const float* __restrict__ w,    // [U]
        float* __restrict__ out) {      // [T*N]
    const int lane = threadIdx.x & 31;
    const int wave = threadIdx.x >> 5;
    const int m    = lane & 15;
    const bool hi  = lane >= 16;

    const int tn0 = (blockIdx.x * WAVES_PER_BLOCK + wave) * OUT_PER_WAVE;

    // Broadcast-weight B tiles, one v2f per K-group of 4 units.
    // B layout (4x16, 2 VGPRs): VGPR0 = rows K0 (lanes 0-15) / K2 (lanes 16-31),
    //                           VGPR1 = rows K1 / K3.
    v2f b[16];
#pragma unroll
    for (int kb = 0; kb < 16; ++kb) {
        const int r0 = 4 * kb + (hi ? 2 : 0);
        b[kb].x = w[r0];
        b[kb].y = w[r0 + 1];
    }

    v8f c0 = {}, c1 = {}, c2 = {}, c3 = {};

    const float* p = X + tn0 + m;
#pragma unroll
    for (int kb = 0; kb < 16; ++kb) {
        const int r0 = 4 * kb + (hi ? 2 : 0);     // A layout mirrors B: VGPR0=K0/K2, VGPR1=K1/K3
        const float* row0 = p + (size_t)r0 * TN_;
        const float* row1 = row0 + (size_t)TN_;
        v2f a0 = { row0[0],  row1[0]  };          // tile 0: tn0 +  0..15
        v2f a1 = { row0[16], row1[16] };          // tile 1: tn0 + 16..31
        v2f a2 = { row0[32], row1[32] };          // tile 2: tn0 + 32..47
        v2f a3 = { row0[48], row1[48] };          // tile 3: tn0 + 48..63
        // 8 args: (neg_a, A, neg_b, B, c_mod, C, reuse_a, reuse_b)
        c0 = __builtin_amdgcn_wmma_f32_16x16x4_f32(false, a0, false, b[kb], (short)0, c0, false, false);
        c1 = __builtin_amdgcn_wmma_f32_16x16x4_f32(false, a1, false, b[kb], (short)0, c1, false, false);
        c2 = __builtin_amdgcn_wmma_f32_16x16x4_f32(false, a2, false, b[kb], (short)0, c2, false, false);
        c3 = __builtin_amdgcn_wmma_f32_16x16x4_f32(false, a3, false, b[kb], (short)0, c3, false, false);
    }

    // D layout (16x16 f32): VGPR r -> M=r on lanes 0-15, M=8+r on lanes 16-31.
    // All N columns are identical, so lane 0 writes outputs 0..7 and lane 16
    // writes outputs 8..15 of each tile, as two b128 stores each.
    if (m == 0) {
        float* o = out + tn0 + (hi ? 8 : 0);
        *(v4f*)(o +  0) = c0.lo;  *(v4f*)(o +  4) = c0.hi;
        *(v4f*)(o + 16) = c1.lo;  *(v4f*)(o + 20) = c1.hi;
        *(v4f*)(o + 32) = c2.lo;  *(v4f*)(o + 36) = c2.hi;
        *(v4f*)(o + 48) = c3.lo;  *(v4f*)(o + 52) = c3.hi;
    }
}

extern "C" void kernel_launch(void* const* d_in, const int* in_sizes, int n_in,
                              void* d_out, int out_size, void* d_ws, size_t ws_size,
                              hipStream_t stream) {
    const float* unit_outputs = (const float*)d_in[0];  // [U, T, N] f32
    const float* conn         = (const float*)d_in[1];  // [U, U] f32
    float* out = (float*)d_out;                         // [T, N] f32
    float* w   = (float*)d_ws;                          // 64 floats of scratch

    amn_weights_kernel<<<1, U_, 0, stream>>>(conn, w);
    amn_wsum_wmma_kernel<<<GRID_BLOCKS, 256, 0, stream>>>(unit_outputs, w, out);
}